// GAT_18949395710230
// MI455X (gfx1250) — compile-verified
//
#include <hip/hip_runtime.h>
#include <math.h>
#include <stdint.h>

typedef __attribute__((ext_vector_type(2))) float    v2f;
typedef __attribute__((ext_vector_type(8))) float    v8f;
typedef __attribute__((ext_vector_type(4))) unsigned v4u;
typedef __attribute__((ext_vector_type(8))) unsigned v8u;

#define NEG_INF_ENC 0x007FFFFFu  // order-encoding of -inf
#define LDS_STRIDE  260          // 256 + 4 pad DWORDs (TDM pad: interval=256dw, amount=4dw)

__device__ __forceinline__ unsigned f2ord(float f) {
    unsigned u = __float_as_uint(f);
    return (u & 0x80000000u) ? ~u : (u | 0x80000000u);
}
__device__ __forceinline__ float ord2f(unsigned e) {
    unsigned u = (e & 0x80000000u) ? (e ^ 0x80000000u) : ~e;
    return __uint_as_float(u);
}
__device__ __forceinline__ float lrelu(float x) { return x > 0.f ? x : 0.2f * x; }

// ---------------------------------------------------------------------------
// H[N,D] = X[N,256] @ W[256,D]   (fp32-exact via V_WMMA_F32_16X16X4_F32)
// Block = 128 threads = 4 waves; each wave owns one 16x16 output tile.
// The 16x256 A-strip is brought into LDS by the Tensor Data Mover:
//   2-group D#: 2D tensor [tensor_dim1=N rows x tensor_dim0=256 cols, f32],
//   tile 16x256, OOB rows zero-filled by TDM, LDS padding 4 DWORDs per
//   256-DWORD row -> row stride 260 (conflict-free 16-row fragment reads).
// Per k-step fragments follow the ISA 16x4 f32 layout:
//   lanes 0-15 -> K=k,k+1 ; lanes 16-31 -> K=k+2,k+3.
// ---------------------------------------------------------------------------
template<int D>
__global__ __launch_bounds__(128)
void gat_gemm_wmma(const float* __restrict__ X, const float* __restrict__ W,
                   float* __restrict__ H, int nrows)
{
    __shared__ float As[16][LDS_STRIDE];
    const int m0   = blockIdx.x * 16;
    const int wave = threadIdx.x >> 5;
    const int lane = threadIdx.x & 31;
    const int half = lane >> 4;
    const int l    = lane & 15;
    const int col0 = blockIdx.y * 64 + wave * 16;

    if (wave == 0) {
        // ---- build Tensor DMA descriptor (uniform; ends up in SGPRs) ----
        unsigned lds_base = (unsigned)(uintptr_t)(&As[0][0]);
        unsigned long long ga =
            (unsigned long long)(uintptr_t)(X + (size_t)m0 * 256);
        v4u g0;
        g0[0] = 1u;                                   // count=1, user descriptor
        g0[1] = lds_base;                             // lds_addr
        g0[2] = (unsigned)ga;                         // global_addr[31:0]
        g0[3] = ((unsigned)(ga >> 32) & 0x01FFFFFFu)  // global_addr[56:32]
                | 0x80000000u;                        // type=2 ("image")
        v8u g1;
        g1[0] = (2u << 16)      // data_size = 4 bytes
              | (1u << 20)      // pad_enable
              | (7u << 22)      // pad_interval: 256 DWORDs (one row)
              | (3u << 25);     // pad_amount: 4 DWORDs
        g1[1] = (256u & 0xFFFFu) << 16;                       // tensor_dim0 lo
        g1[2] = (256u >> 16) | (((unsigned)nrows & 0xFFFFu) << 16); // dim0 hi | dim1 lo
        g1[3] = ((unsigned)nrows >> 16) | (256u << 16);       // dim1 hi | tile_dim0=256
        g1[4] = 16u;                                          // tile_dim1=16, tile_dim2=0
        g1[5] = 256u;                                         // tensor_dim0_stride lo32
        g1[6] = 0u;                                           // stride0 hi | stride1 lo
        g1[7] = 0u;
        asm volatile("tensor_load_to_lds %0, %1" :: "s"(g0), "s"(g1) : "memory");
        __builtin_amdgcn_s_wait_tensorcnt(0);
    }
    __syncthreads();

    v8f acc = {};
    #pragma unroll 8
    for (int k = 0; k < 256; k += 4) {
        v2f a, b;
        a.x = As[l][k + 2 * half + 0];
        a.y = As[l][k + 2 * half + 1];
        const float* wp = W + (size_t)(k + 2 * half) * D + col0 + l;
        b.x = wp[0];
        b.y = wp[D];
        acc = __builtin_amdgcn_wmma_f32_16x16x4_f32(false, a, false, b,
                                                    (short)0, acc, false, false);
    }

    // C/D layout: VGPR r -> M=r (lanes 0-15), M=r+8 (lanes 16-31)
    const int col = col0 + l;
    float* hp = H + (size_t)(m0 + 8 * half) * D + col;
    if (m0 + 16 <= nrows) {               // block-uniform fast path
        #pragma unroll
        for (int r = 0; r < 8; ++r) hp[(size_t)r * D] = acc[r];
    } else {
        #pragma unroll
        for (int r = 0; r < 8; ++r)
            if (m0 + 8 * half + r < nrows) hp[(size_t)r * D] = acc[r];
    }
}

// ---------------------------------------------------------------------------
// s_src[i] = h[i] . a[0:d], s_dst[i] = h[i] . a[d:2d]   (wave per node)
// ---------------------------------------------------------------------------
__global__ __launch_bounds__(256)
void gat_node_scores(const float* __restrict__ H, const float* __restrict__ a,
                     float* __restrict__ s_src, float* __restrict__ s_dst,
                     int n, int d)
{
    int wave = threadIdx.x >> 5, lane = threadIdx.x & 31;
    int node = blockIdx.x * 8 + wave;
    if (node >= n) return;
    const float* hr = H + (size_t)node * d;
    float ss = 0.f, sd = 0.f;
    for (int j = lane; j < d; j += 32) {
        float hv = hr[j];
        ss += hv * a[j];
        sd += hv * a[d + j];
    }
    for (int off = 16; off; off >>= 1) {
        ss += __shfl_down(ss, off, 32);
        sd += __shfl_down(sd, off, 32);
    }
    if (lane == 0) { s_src[node] = ss; s_dst[node] = sd; }
}

// ---------------------------------------------------------------------------
__global__ void gat_init_layer(unsigned* __restrict__ m_enc, float* __restrict__ z,
                               float* __restrict__ out, int n, long long outElems)
{
    long long i = (long long)blockIdx.x * blockDim.x + threadIdx.x;
    if (i < n) { m_enc[i] = NEG_INF_ENC; z[i] = 0.f; }
    if (i < outElems) out[i] = 0.f;
}

__global__ void gat_edge_max(const int* __restrict__ src, const int* __restrict__ dst,
                             const float* __restrict__ s_src, const float* __restrict__ s_dst,
                             unsigned* __restrict__ m_enc, long long E)
{
    long long e = (long long)blockIdx.x * blockDim.x + threadIdx.x;
    if (e >= E) return;
    int s = src[e], t = dst[e];
    float sc = lrelu(s_src[s] + s_dst[t]);
    atomicMax(&m_enc[s], f2ord(sc));
}

__global__ void gat_edge_sum(const int* __restrict__ src, const int* __restrict__ dst,
                             const float* __restrict__ s_src, const float* __restrict__ s_dst,
                             const unsigned* __restrict__ m_enc, float* __restrict__ z,
                             long long E)
{
    long long e = (long long)blockIdx.x * blockDim.x + threadIdx.x;
    if (e >= E) return;
    int s = src[e], t = dst[e];
    float sc = lrelu(s_src[s] + s_dst[t]);
    atomicAdd(&z[s], expf(sc - ord2f(m_enc[s])));
}

// Wave per edge: out[src] += attn * h[dst]  (L2-resident GLOBAL_ATOMIC_ADD_F32)
template<int D>
__global__ __launch_bounds__(256)
void gat_edge_aggregate(const int* __restrict__ src, const int* __restrict__ dst,
                        const float* __restrict__ s_src, const float* __restrict__ s_dst,
                        const unsigned* __restrict__ m_enc, const float* __restrict__ z,
                        const float* __restrict__ H, float* __restrict__ out,
                        long long E)
{
    long long e = (long long)blockIdx.x * 8 + (threadIdx.x >> 5);
    if (e >= E) return;
    int lane = threadIdx.x & 31;
    int s = src[e], t = dst[e];
    float sc = lrelu(s_src[s] + s_dst[t]);
    float attn = expf(sc - ord2f(m_enc[s])) / z[s];
    const float* hv = H + (size_t)t * D + lane;
    float* ov = out + (size_t)s * D + lane;
    #pragma unroll
    for (int j = 0; j < D / 32; ++j)
        atomicAdd(&ov[32 * j], attn * hv[32 * j]);
}

__global__ void gat_elu_inplace(float* __restrict__ x, long long nElems)
{
    long long i = (long long)blockIdx.x * blockDim.x + threadIdx.x;
    if (i >= nElems) return;
    float v = x[i];
    x[i] = v > 0.f ? v : (expf(v) - 1.f);
}

// ---------------------------------------------------------------------------
// Head: logits = h1 @ lin_w + lin_b ; out = log_softmax(logits)
// One block (64 threads) per node.
// ---------------------------------------------------------------------------
__global__ __launch_bounds__(64)
void gat_head_logsoftmax(const float* __restrict__ H, const float* __restrict__ Wl,
                         const float* __restrict__ bl, float* __restrict__ out, int n)
{
    __shared__ float hrow[128];
    __shared__ float lg[40];
    __shared__ float s_lse;
    int node = blockIdx.x;
    if (node >= n) return;
    int t = threadIdx.x;
    hrow[t]      = H[(size_t)node * 128 + t];
    hrow[t + 64] = H[(size_t)node * 128 + t + 64];
    __syncthreads();
    if (t < 40) {
        float acc = bl[t];
        #pragma unroll 8
        for (int k = 0; k < 128; ++k) acc += hrow[k] * Wl[k * 40 + t];
        lg[t] = acc;
    }
    __syncthreads();
    if (t < 32) {
        float v0 = lg[t];                                  // t < 32 < 40
        float v1 = (t + 32 < 40) ? lg[t + 32] : -INFINITY;
        float m = fmaxf(v0, v1);
        for (int off = 16; off; off >>= 1) m = fmaxf(m, __shfl_xor(m, off, 32));
        float sum = expf(v0 - m) + ((t + 32 < 40) ? expf(v1 - m) : 0.f);
        for (int off = 16; off; off >>= 1) sum += __shfl_xor(sum, off, 32);
        if (t == 0) s_lse = m + logf(sum);
    }
    __syncthreads();
    if (t < 40) out[(size_t)node * 40 + t] = lg[t] - s_lse;
}

// ---------------------------------------------------------------------------
extern "C" void kernel_launch(void* const* d_in, const int* in_sizes, int n_in,
                              void* d_out, int out_size, void* d_ws, size_t ws_size,
                              hipStream_t stream)
{
    (void)n_in; (void)out_size; (void)ws_size;
    const float* x     = (const float*)d_in[0];
    const int*   edge  = (const int*)d_in[1];
    const float* W0    = (const float*)d_in[2];
    const float* a0    = (const float*)d_in[3];
    const float* W1    = (const float*)d_in[4];
    const float* a1    = (const float*)d_in[5];
    const float* lin_w = (const float*)d_in[6];
    const float* lin_b = (const float*)d_in[7];

    const int       N = in_sizes[0] / 256;
    const long long E = in_sizes[1] / 2;
    const int* src = edge;
    const int* dst = edge + E;

    // workspace layout
    char* ws = (char*)d_ws;
    const size_t szH = (size_t)N * 256 * sizeof(float);
    float*    bufA = (float*)(ws);             // h of current layer
    float*    bufB = (float*)(ws + szH);       // layer input / aggregate
    float*    ssrc = (float*)(ws + 2 * szH);
    float*    sdst = (float*)(ws + 2 * szH + (size_t)N * 4);
    unsigned* menc = (unsigned*)(ws + 2 * szH + (size_t)N * 8);
    float*    zbuf = (float*)(ws + 2 * szH + (size_t)N * 12);

    const int mtiles = (N + 15) / 16;
    const int eblk   = (int)((E + 255) / 256);
    const int ewave  = (int)((E + 7) / 8);

    // ---- layer 0 (256 -> 256) ----
    gat_gemm_wmma<256><<<dim3(mtiles, 4), 128, 0, stream>>>(x, W0, bufA, N);
    gat_node_scores<<<(N + 7) / 8, 256, 0, stream>>>(bufA, a0, ssrc, sdst, N, 256);
    {
        long long oe = (long long)N * 256;
        gat_init_layer<<<(unsigned)((oe + 255) / 256), 256, 0, stream>>>(menc, zbuf, bufB, N, oe);
    }
    gat_edge_max<<<eblk, 256, 0, stream>>>(src, dst, ssrc, sdst, menc, E);
    gat_edge_sum<<<eblk, 256, 0, stream>>>(src, dst, ssrc, sdst, menc, zbuf, E);
    gat_edge_aggregate<256><<<ewave, 256, 0, stream>>>(src, dst, ssrc, sdst, menc, zbuf,
                                                       bufA, bufB, E);
    gat_elu_inplace<<<(unsigned)(((long long)N * 256 + 255) / 256), 256, 0, stream>>>(
        bufB, (long long)N * 256);

    // ---- layer 1 (256 -> 128) ----
    gat_gemm_wmma<128><<<dim3(mtiles, 2), 128, 0, stream>>>(bufB, W1, bufA, N);
    gat_node_scores<<<(N + 7) / 8, 256, 0, stream>>>(bufA, a1, ssrc, sdst, N, 128);
    {
        long long oe = (long long)N * 128;
        gat_init_layer<<<(unsigned)((oe + 255) / 256), 256, 0, stream>>>(menc, zbuf, bufB, N, oe);
    }
    gat_edge_max<<<eblk, 256, 0, stream>>>(src, dst, ssrc, sdst, menc, E);
    gat_edge_sum<<<eblk, 256, 0, stream>>>(src, dst, ssrc, sdst, menc, zbuf, E);
    gat_edge_aggregate<128><<<ewave, 256, 0, stream>>>(src, dst, ssrc, sdst, menc, zbuf,
                                                       bufA, bufB, E);
    gat_elu_inplace<<<(unsigned)(((long long)N * 128 + 255) / 256), 256, 0, stream>>>(
        bufB, (long long)N * 128);

    // ---- head ----
    gat_head_logsoftmax<<<N, 64, 0, stream>>>(bufB, lin_w, lin_b, (float*)d_out, N);
}